// VectorQuantizer_39058432590352
// MI455X (gfx1250) — compile-verified
//
#include <hip/hip_runtime.h>
#include <hip/hip_bf16.h>
#include <cfloat>

typedef __attribute__((ext_vector_type(2))) float v2f;
typedef __attribute__((ext_vector_type(8))) float v8f;

#define B_SZ 65536
#define D_SZ 128
#define K_SZ 4096
#define KT   32          // codebook rows per LDS buffer (double-buffered)
#define LDS_STRIDE 132   // padded floats per row: conflict-free banks, 16B aligned
#define ROWS_PER_BLOCK 128

#if __has_builtin(__builtin_amdgcn_global_load_async_to_lds_b128)
#define USE_ASYNC_LDS 1
// Pointee type must exactly match the builtin's parameter pointee:
//   __attribute__((__vector_size__(4 * sizeof(int)))) int
typedef int vsi4 __attribute__((vector_size(16)));
typedef vsi4 __attribute__((address_space(1)))* g_vsi4_ptr;   // global (AS1)
typedef vsi4 __attribute__((address_space(3)))* l_vsi4_ptr;   // LDS (AS3)

__device__ __forceinline__ void async_copy16_to_lds(const void* gsrc, void* ldst) {
    __builtin_amdgcn_global_load_async_to_lds_b128(
        (g_vsi4_ptr)(unsigned long long)gsrc,
        (l_vsi4_ptr)(unsigned int)(unsigned long long)ldst,
        0, 0);
}

__device__ __forceinline__ void wait_async_lds() {
#if __has_builtin(__builtin_amdgcn_s_wait_asynccnt)
    __builtin_amdgcn_s_wait_asynccnt(0);
#else
    asm volatile("s_wait_asynccnt 0x0" ::: "memory");
#endif
}
#else
#define USE_ASYNC_LDS 0
#endif

// ---------------------------------------------------------------- e2[k] = ||e_k||^2
__global__ __launch_bounds__(256)
void vq_e2_kernel(const float* __restrict__ emb, float* __restrict__ e2) {
    int k = blockIdx.x * blockDim.x + threadIdx.x;
    if (k >= K_SZ) return;
    const float4* row = (const float4*)(emb + (size_t)k * D_SZ);
    float s = 0.f;
#pragma unroll
    for (int i = 0; i < D_SZ / 4; ++i) {
        float4 v = row[i];
        s += v.x * v.x + v.y * v.y + v.z * v.z + v.w * v.w;
    }
    e2[k] = s;
}

// ---------------------------------------------------------------- fused GEMM + argmin
// score[b,k] = e2[k] - 2 * (x_b . e_k)   (x2 per-row constant -> irrelevant to argmin)
__global__ __launch_bounds__(256, 1)
void vq_argmin_kernel(const float* __restrict__ enc,
                      const float* __restrict__ emb,
                      const float* __restrict__ e2,
                      float* __restrict__ out_idx_f,
                      int* __restrict__ out_idx_i) {
    __shared__ float lds_e[2][KT * LDS_STRIDE];   // double buffer, ~2 x 16.9 KB

    const int tid  = threadIdx.x;
    const int lane = tid & 31;
    const int wave = tid >> 5;
    const int n    = lane & 15;   // A: row-in-tile, B: col-in-tile, C: N column
    const int half = lane >> 4;   // selects K-pair {0,1} vs {2,3} of the k=4 slice

    const int row0 = blockIdx.x * ROWS_PER_BLOCK + wave * 16;

    // Preload this wave's 16x128 A tile into VGPRs.
    // 32-bit A 16x4 layout: lanes 0-15 hold K={0,1}, lanes 16-31 hold K={2,3}.
    v2f areg[32];
    {
        const float* aptr = enc + (size_t)(row0 + n) * D_SZ + 2 * half;
#pragma unroll
        for (int dt = 0; dt < 32; ++dt) {
            areg[dt].x = aptr[dt * 4 + 0];
            areg[dt].y = aptr[dt * 4 + 1];
        }
    }

    float minv[8];
    int   mini[8];
#pragma unroll
    for (int i = 0; i < 8; ++i) { minv[i] = FLT_MAX; mini[i] = 0; }

    // Per-thread staging map: 4 x float4 per thread covers KT*32 float4 per chunk.
    const int sr[4] = { (tid + 0 * 256) >> 5, (tid + 1 * 256) >> 5,
                        (tid + 2 * 256) >> 5, (tid + 3 * 256) >> 5 };
    const int sc = tid & 31;   // float4 column (same for all 4 pieces)

    const int nchunks = K_SZ / KT;   // 128

    // ---- prologue: stage chunk 0 into buffer 0
    {
        const float4* src = (const float4*)emb;   // chunk 0
#if USE_ASYNC_LDS
#pragma unroll
        for (int j = 0; j < 4; ++j)
            async_copy16_to_lds(&src[sr[j] * (D_SZ / 4) + sc],
                                &lds_e[0][sr[j] * LDS_STRIDE + sc * 4]);
        wait_async_lds();
#else
#pragma unroll
        for (int j = 0; j < 4; ++j) {
            float4 v = src[sr[j] * (D_SZ / 4) + sc];
            *(float4*)&lds_e[0][sr[j] * LDS_STRIDE + sc * 4] = v;
        }
#endif
    }
    __syncthreads();

    for (int c = 0; c < nchunks; ++c) {
        const int kc = c * KT;
        const float* cur = lds_e[c & 1];
        float*       nxt = lds_e[(c + 1) & 1];
        const bool have_next = (c + 1 < nchunks);

#if USE_ASYNC_LDS
        // Kick off next chunk's global->LDS DMA before computing (overlaps).
        if (have_next) {
            const float4* src = (const float4*)(emb + (size_t)(kc + KT) * D_SZ);
#pragma unroll
            for (int j = 0; j < 4; ++j)
                async_copy16_to_lds(&src[sr[j] * (D_SZ / 4) + sc],
                                    &nxt[sr[j] * LDS_STRIDE + sc * 4]);
        }
#else
        // Fallback: pull next chunk into registers now, spill to LDS after compute.
        float4 t[4];
        if (have_next) {
            const float4* src = (const float4*)(emb + (size_t)(kc + KT) * D_SZ);
#pragma unroll
            for (int j = 0; j < 4; ++j) t[j] = src[sr[j] * (D_SZ / 4) + sc];
        }
#endif

        // ---- compute: two 16x16 N-tiles with independent accumulators (ILP).
        // B (4x16, f32): lanes 0-15 hold rows K={0,1}, lanes 16-31 rows K={2,3}.
        const float* b0 = cur + (0 * 16 + n) * LDS_STRIDE + 2 * half;
        const float* b1 = cur + (1 * 16 + n) * LDS_STRIDE + 2 * half;
        v8f acc0 = {0.f, 0.f, 0.f, 0.f, 0.f, 0.f, 0.f, 0.f};
        v8f acc1 = {0.f, 0.f, 0.f, 0.f, 0.f, 0.f, 0.f, 0.f};
#pragma unroll
        for (int dt = 0; dt < 32; ++dt) {
            v2f bb0 = *(const v2f*)(b0 + dt * 4);
            v2f bb1 = *(const v2f*)(b1 + dt * 4);
            acc0 = __builtin_amdgcn_wmma_f32_16x16x4_f32(
                     false, areg[dt], false, bb0, (short)0, acc0, false, false);
            acc1 = __builtin_amdgcn_wmma_f32_16x16x4_f32(
                     false, areg[dt], false, bb1, (short)0, acc1, false, false);
        }

        // C layout: VGPR i = rows {i, i+8}, lane&15 = column N
        const float e2v0 = e2[kc + n];
        const float e2v1 = e2[kc + 16 + n];
        const int   k0   = kc + n;
        const int   k1   = kc + 16 + n;
#pragma unroll
        for (int i = 0; i < 8; ++i) {
            float v0 = e2v0 - 2.0f * acc0[i];
            float v1 = e2v1 - 2.0f * acc1[i];
            if (v0 < minv[i]) { minv[i] = v0; mini[i] = k0; }
            if (v1 < minv[i]) { minv[i] = v1; mini[i] = k1; }
        }

#if USE_ASYNC_LDS
        if (have_next) wait_async_lds();
#else
        if (have_next) {
#pragma unroll
            for (int j = 0; j < 4; ++j)
                *(float4*)&nxt[sr[j] * LDS_STRIDE + sc * 4] = t[j];
        }
#endif
        __syncthreads();
    }

    // Min+index reduction across the 16 lanes sharing each output row.
#pragma unroll
    for (int i = 0; i < 8; ++i) {
        float v  = minv[i];
        int   ix = mini[i];
#pragma unroll
        for (int m = 8; m >= 1; m >>= 1) {
            float ov = __shfl_xor(v, m, 32);
            int   oi = __shfl_xor(ix, m, 32);
            if (ov < v || (ov == v && oi < ix)) { v = ov; ix = oi; }  // first-min tie-break
        }
        if (n == 0) {
            int row = row0 + i + 8 * half;
            out_idx_f[row] = (float)ix;
            out_idx_i[row] = ix;
        }
    }
}

// ---------------------------------------------------------------- gather + ST output + MSE partial sums
__global__ __launch_bounds__(256)
void vq_gather_loss_kernel(const float* __restrict__ enc,
                           const float* __restrict__ emb,
                           const int* __restrict__ idx,
                           float* __restrict__ out_st,
                           double* __restrict__ acc) {
    __shared__ double wsum[8];
    int g = blockIdx.x * 256 + threadIdx.x;     // one float4 per thread
    int b = g >> 5;
    int c = (g & 31) * 4;
    int k = idx[b];
    float4 x4 = *(const float4*)(enc + (size_t)b * D_SZ + c);
    float4 q4 = *(const float4*)(emb + (size_t)k * D_SZ + c);
    float4 st;
    st.x = x4.x + (q4.x - x4.x);
    st.y = x4.y + (q4.y - x4.y);
    st.z = x4.z + (q4.z - x4.z);
    st.w = x4.w + (q4.w - x4.w);
    *(float4*)(out_st + (size_t)b * D_SZ + c) = st;

    float dx = x4.x - q4.x, dy = x4.y - q4.y, dz = x4.z - q4.z, dw = x4.w - q4.w;
    double s = (double)(dx * dx + dy * dy) + (double)(dz * dz + dw * dw);
#pragma unroll
    for (int m = 16; m >= 1; m >>= 1) s += __shfl_xor(s, m, 32);
    if ((threadIdx.x & 31) == 0) wsum[threadIdx.x >> 5] = s;
    __syncthreads();
    if (threadIdx.x == 0) {
        double t = 0.0;
#pragma unroll
        for (int i = 0; i < 8; ++i) t += wsum[i];
        atomicAdd(acc, t);   // global_atomic_add_f64
    }
}

// ---------------------------------------------------------------- scalar losses
__global__ void vq_finalize_kernel(const double* __restrict__ acc,
                                   float* __restrict__ out_scalars) {
    double mse = *acc / (double)((size_t)B_SZ * (size_t)D_SZ);
    out_scalars[0] = (float)(mse * 0.25 + mse); // vq_loss = BETA*commitment + embedding
    out_scalars[1] = (float)mse;                // embedding_loss
    out_scalars[2] = (float)mse;                // commitment_loss
}

// ----------------------------------------------------------------
extern "C" void kernel_launch(void* const* d_in, const int* in_sizes, int n_in,
                              void* d_out, int out_size, void* d_ws, size_t ws_size,
                              hipStream_t stream) {
    const float* enc = (const float*)d_in[0];   // (B, D) fp32
    const float* emb = (const float*)d_in[1];   // (K, D) fp32
    float* out = (float*)d_out;                 // [idx(B) | st(B*D) | vq | emb_loss | commit]

    float*  ws_e2  = (float*)d_ws;                                    // K floats
    int*    ws_idx = (int*)((char*)d_ws + (size_t)K_SZ * 4);          // B ints
    double* ws_acc = (double*)((char*)d_ws + (size_t)K_SZ * 4 + (size_t)B_SZ * 4);

    (void)hipMemsetAsync(ws_acc, 0, sizeof(double), stream);

    vq_e2_kernel<<<K_SZ / 256, 256, 0, stream>>>(emb, ws_e2);
    vq_argmin_kernel<<<B_SZ / ROWS_PER_BLOCK, 256, 0, stream>>>(enc, emb, ws_e2, out, ws_idx);
    vq_gather_loss_kernel<<<(B_SZ * (D_SZ / 4)) / 256, 256, 0, stream>>>(
        enc, emb, ws_idx, out + B_SZ, ws_acc);
    vq_finalize_kernel<<<1, 1, 0, stream>>>(ws_acc, out + B_SZ + (size_t)B_SZ * D_SZ);
}